// SeqOuterProductMean_34660386079026
// MI455X (gfx1250) — compile-verified
//
#include <hip/hip_runtime.h>

typedef __attribute__((ext_vector_type(2))) float v2f;
typedef __attribute__((ext_vector_type(8))) float v8f;

#define SEQ_DIM 256
#define PAIR_DIM 128
#define HIDDEN 32
#define LSEQ 384
#define EPSV 1e-5f

// ---------------------------------------------------------------------------
// Kernel 1: LayerNorm + the two H=32 projections. One block per sequence row.
// ---------------------------------------------------------------------------
__global__ __launch_bounds__(256) void ln_proj_kernel(
    const float* __restrict__ seq,
    const float* __restrict__ gamma, const float* __restrict__ beta,
    const float* __restrict__ w1, const float* __restrict__ b1,
    const float* __restrict__ w2, const float* __restrict__ b2,
    float* __restrict__ s1, float* __restrict__ s2)
{
    __shared__ float xs[SEQ_DIM];
    __shared__ float red[256];
    __shared__ float s_mu, s_rstd;

    const int r = blockIdx.x;
    const int tid = threadIdx.x;

    float v = seq[r * SEQ_DIM + tid];

    // mean
    red[tid] = v;
    __syncthreads();
    for (int off = 128; off > 0; off >>= 1) {
        if (tid < off) red[tid] += red[tid + off];
        __syncthreads();
    }
    if (tid == 0) s_mu = red[0] * (1.0f / SEQ_DIM);
    __syncthreads();
    const float mu = s_mu;
    const float d = v - mu;

    // variance
    red[tid] = d * d;
    __syncthreads();
    for (int off = 128; off > 0; off >>= 1) {
        if (tid < off) red[tid] += red[tid + off];
        __syncthreads();
    }
    if (tid == 0) s_rstd = rsqrtf(red[0] * (1.0f / SEQ_DIM) + EPSV);
    __syncthreads();

    xs[tid] = d * s_rstd * gamma[tid] + beta[tid];
    __syncthreads();

    // 64 dot products of length 256 (s1: h<32, s2: h>=32)
    if (tid < 2 * HIDDEN) {
        const int h = tid & (HIDDEN - 1);
        const bool first = tid < HIDDEN;
        const float* w = (first ? w1 : w2) + h * SEQ_DIM;
        float acc = first ? b1[h] : b2[h];
        #pragma unroll 8
        for (int dd = 0; dd < SEQ_DIM; ++dd) acc += xs[dd] * w[dd];
        (first ? s1 : s2)[r * HIDDEN + h] = acc;
    }
}

// ---------------------------------------------------------------------------
// Kernel 2: t[i, p*32+f] = sum_d s1[i,d] * w3[p, d*32+f]
// GEMM M=384, N=4096, K=32 via V_WMMA_F32_16X16X4_F32; one 16x16 tile / wave.
// Stores are lane-contiguous (n fastest) -> coalesced.
// ---------------------------------------------------------------------------
__global__ __launch_bounds__(256) void t_gemm_kernel(
    const float* __restrict__ s1, const float* __restrict__ w3,
    float* __restrict__ t)
{
    const int tid  = threadIdx.x;
    const int wave = tid >> 5;
    const int lane = tid & 31;

    const int task   = blockIdx.x * 8 + wave;  // 24 * 256 = 6144 tasks
    const int i_tile = task >> 8;              // 0..23
    const int n_tile = task & 255;             // 0..255

    const int m    = lane & 15;                // A row within tile
    const int kb   = (lane >> 4) << 1;         // K sub-pair: {0,1} or {2,3}
    const int nloc = lane & 15;                // B/D column within tile
    const int n    = n_tile * 16 + nloc;       // global column 0..4095
    const int p    = n >> 5;
    const int f    = n & 31;

    const float* arow = s1 + (i_tile * 16 + m) * HIDDEN;
    const float* bcol = w3 + p * (HIDDEN * HIDDEN) + f;   // stride HIDDEN per k

    v8f c = {};
    #pragma unroll
    for (int kk = 0; kk < 8; ++kk) {
        v2f a, b;
        a.x = arow[kk * 4 + kb];
        a.y = arow[kk * 4 + kb + 1];
        b.x = bcol[(kk * 4 + kb) * HIDDEN];
        b.y = bcol[(kk * 4 + kb + 1) * HIDDEN];
        c = __builtin_amdgcn_wmma_f32_16x16x4_f32(
                false, a, false, b, (short)0, c, false, false);
    }

    const int mbase = (lane >> 4) << 3;        // rows 0..7 or 8..15
    #pragma unroll
    for (int rr = 0; rr < 8; ++rr) {
        t[(i_tile * 16 + mbase + rr) * (PAIR_DIM * HIDDEN) + n] = c[rr];
    }
}

// ---------------------------------------------------------------------------
// Kernel 3: pair[i,j,p] = sum_f s2[j,f] * t[i,p,f] + b3[p]
// Per fixed i this is a GEMM with A = s2 (j x f), B = t[i]^T (f x p), so the
// WMMA N dimension is p (the fastest output axis) -> coalesced 64B stores.
// Block = one i (8 waves = 8 p-tiles covering all 128 p); each wave holds its
// B fragments (t) in VGPRs across all 24 j-tiles; s2 staged in LDS (48 KB).
// ---------------------------------------------------------------------------
__global__ __launch_bounds__(256) void pair_gemm_kernel(
    const float* __restrict__ t, const float* __restrict__ s2,
    const float* __restrict__ b3, float* __restrict__ out)
{
    __shared__ float s2_lds[LSEQ * HIDDEN];    // 48 KB

    const int tid = threadIdx.x;
    for (int idx = tid; idx < LSEQ * HIDDEN; idx += 256)
        s2_lds[idx] = s2[idx];
    __syncthreads();

    const int wave = tid >> 5;                 // p-tile 0..7
    const int lane = tid & 31;
    const int i    = blockIdx.x;               // 0..383

    const int m     = lane & 15;               // A row (j within tile)
    const int kb    = (lane >> 4) << 1;        // K sub-pair
    const int nloc  = lane & 15;               // D col (p within tile)
    const int mbase = (lane >> 4) << 3;        // D rows 0..7 / 8..15
    const int p     = wave * 16 + nloc;        // this lane's output p

    // B fragments: B[k][p] = t[i, p, k]  (held in VGPRs for all 24 j-tiles)
    const float* tcol = t + ((size_t)i * PAIR_DIM + p) * HIDDEN;
    v2f bfrag[8];
    #pragma unroll
    for (int kk = 0; kk < 8; ++kk) {
        bfrag[kk].x = tcol[kk * 4 + kb];
        bfrag[kk].y = tcol[kk * 4 + kb + 1];
    }
    const float bias = b3[p];

    for (int j_tile = 0; j_tile < LSEQ / 16; ++j_tile) {
        const float* arow = s2_lds + (j_tile * 16 + m) * HIDDEN;
        v8f c = {};
        #pragma unroll
        for (int kk = 0; kk < 8; ++kk) {
            v2f a;
            a.x = arow[kk * 4 + kb];
            a.y = arow[kk * 4 + kb + 1];
            c = __builtin_amdgcn_wmma_f32_16x16x4_f32(
                    false, a, false, bfrag[kk], (short)0, c, false, false);
        }
        // D row rr maps to j = j_tile*16 + mbase + rr; lanes cover 16
        // consecutive p -> each store = 2 contiguous 64B cacheline segments.
        float* orow = out + ((size_t)i * LSEQ + j_tile * 16 + mbase) * PAIR_DIM
                          + wave * 16 + nloc;
        #pragma unroll
        for (int rr = 0; rr < 8; ++rr) {
            orow[(size_t)rr * PAIR_DIM] = c[rr] + bias;
        }
    }
}

// ---------------------------------------------------------------------------
extern "C" void kernel_launch(void* const* d_in, const int* in_sizes, int n_in,
                              void* d_out, int out_size, void* d_ws, size_t ws_size,
                              hipStream_t stream) {
    const float* seq   = (const float*)d_in[0];
    const float* gamma = (const float*)d_in[1];
    const float* beta  = (const float*)d_in[2];
    const float* w1    = (const float*)d_in[3];
    const float* b1    = (const float*)d_in[4];
    const float* w2    = (const float*)d_in[5];
    const float* b2    = (const float*)d_in[6];
    const float* w3    = (const float*)d_in[7];
    const float* b3    = (const float*)d_in[8];

    float* ws = (float*)d_ws;
    float* t  = ws;                                    // 384 * 4096 floats
    float* s1 = ws + (size_t)LSEQ * PAIR_DIM * HIDDEN; // 384 * 32
    float* s2 = s1 + (size_t)LSEQ * HIDDEN;            // 384 * 32
    float* out = (float*)d_out;

    ln_proj_kernel<<<LSEQ, 256, 0, stream>>>(seq, gamma, beta, w1, b1, w2, b2, s1, s2);
    t_gemm_kernel<<<(24 * 256) / 8, 256, 0, stream>>>(s1, w3, t);
    pair_gemm_kernel<<<LSEQ, 256, 0, stream>>>(t, s2, b3, out);
}